// MLCriterion_47090021433792
// MI455X (gfx1250) — compile-verified
//
#include <hip/hip_runtime.h>
#include <hip/hip_bf16.h>
#include <math.h>

#define VOCAB   50000
#define SRC_L   400
#define NROWS   2048
#define PAD_ID  1
#define BLK     256

typedef float f32x4 __attribute__((ext_vector_type(4)));
typedef int   v4i   __attribute__((ext_vector_type(4)));

// Address-space-qualified pointee types (typedefs reliably keep the AS qualifier)
typedef __attribute__((address_space(1))) v4i v4i_g;   // global
typedef __attribute__((address_space(3))) v4i v4i_l;   // LDS

// ---------- CDNA5 async global->LDS (guarded; fallback = plain copy) ----------
#if defined(__has_builtin)
#  if __has_builtin(__builtin_amdgcn_global_load_async_to_lds_b128)
#    define HAVE_ASYNC_B128 1
#  endif
#endif
#ifndef HAVE_ASYNC_B128
#  define HAVE_ASYNC_B128 0
#endif

__device__ __forceinline__ void async_g2l_b128(const void* g, void* lds_generic) {
#if HAVE_ASYNC_B128
    // global pointer -> addrspace(1); LDS generic pointer: flat LDS addr keeps the
    // LDS offset in the low 32 bits (aperture in [63:32]), so truncation recovers AS3.
    v4i_g* gp = (v4i_g*)(unsigned long long)g;
    v4i_l* lp = (v4i_l*)(unsigned int)(unsigned long long)lds_generic;
    __builtin_amdgcn_global_load_async_to_lds_b128(gp, lp, 0, 0);
#else
    *(f32x4*)lds_generic = *(const f32x4*)g;
#endif
}

__device__ __forceinline__ void wait_async0() {
#if HAVE_ASYNC_B128
#  if defined(__has_builtin) && __has_builtin(__builtin_amdgcn_s_wait_asynccnt)
    __builtin_amdgcn_s_wait_asynccnt(0);
#  else
    asm volatile("s_wait_asynccnt 0" ::: "memory");
#  endif
#endif
}

// ---------------- Kernel A: one workgroup per row ----------------
__global__ __launch_bounds__(BLK) void ptrgen_row_kernel(
    const float* __restrict__ p_gen,     // [NROWS, VOCAB]
    const float* __restrict__ p_copy,    // [NROWS, SRC_L]
    const float* __restrict__ p_switch,  // [NROWS]
    const int*   __restrict__ tgt,       // [NROWS]
    const int*   __restrict__ src,       // [NROWS, SRC_L]
    float*       __restrict__ out,       // [1 + NROWS]: [0]=loss (kernel B), [1+r]=pred
    float*       __restrict__ loss_ws)   // [NROWS] per-row loss terms
{
    const int row = blockIdx.x;
    const int t   = threadIdx.x;

    __shared__ __align__(16) float s_pcopy[SRC_L];
    __shared__ __align__(16) int   s_src[SRC_L];
    __shared__ float s_val[BLK];
    __shared__ int   s_idx[BLK];
    __shared__ float s_sum[BLK];

    // ---- Stage p_copy/src rows into LDS via async b128 loads (100 each) ----
    const float* gpc = p_copy + (size_t)row * SRC_L;
    const int*   gps = src    + (size_t)row * SRC_L;
    if (t < SRC_L / 4) {
        async_g2l_b128(gpc + 4 * t, &s_pcopy[4 * t]);
        async_g2l_b128(gps + 4 * t, &s_src[4 * t]);
    }

    const float sw  = p_switch[row];
    const float inv = 1.0f - sw;          // > 0 since sw ~ U[0,1)
    const int   tg  = tgt[row];

    // ---- Dense argmax over p_gen row: NT-hint b128 streaming (read-once, > L2) ----
    const f32x4* grow = (const f32x4*)(p_gen + (size_t)row * VOCAB);
    float bmax = -INFINITY;
    int   bidx = 0;
    const int NCH = VOCAB / 4;            // 12500 float4 chunks, 16B-aligned
    for (int c = t; c < NCH; c += BLK) {
        if (c + BLK < NCH) __builtin_prefetch(&grow[c + BLK], 0, 1); // global_prefetch_b8
        f32x4 v = __builtin_nontemporal_load(&grow[c]);
        const int base = 4 * c;
        if (v.x > bmax) { bmax = v.x; bidx = base;     }
        if (v.y > bmax) { bmax = v.y; bidx = base + 1; }
        if (v.z > bmax) { bmax = v.z; bidx = base + 2; }
        if (v.w > bmax) { bmax = v.w; bidx = base + 3; }
    }
    float val = bmax * inv;               // positive scale preserves argmax
    int   idx = bidx;

    wait_async0();
    __syncthreads();

    // ---- Copy-position candidates (full score) + target copy-sum partial ----
    // Copy additions are >= 0, so true argmax = max(dense-only max, copied full scores).
    float csum_tgt = 0.0f;
    for (int j = t; j < SRC_L; j += BLK) {
        const int v = s_src[j];
        float cs = 0.0f;                  // dedup: sum p_copy over equal src entries
        for (int k = 0; k < SRC_L; ++k)
            cs += (s_src[k] == v) ? s_pcopy[k] : 0.0f;
        const float cand = p_gen[(size_t)row * VOCAB + v] * inv + sw * cs;
        if (cand > val || (cand == val && v < idx)) { val = cand; idx = v; }
        csum_tgt += (v == tg) ? s_pcopy[j] : 0.0f;
    }

    // ---- wave32-safe block reduction: (max,arg) with first-index tie-break + sum ----
    s_val[t] = val; s_idx[t] = idx; s_sum[t] = csum_tgt;
    __syncthreads();
    for (int off = BLK / 2; off > 0; off >>= 1) {
        if (t < off) {
            const float ov = s_val[t + off];
            const int   oi = s_idx[t + off];
            if (ov > s_val[t] || (ov == s_val[t] && oi < s_idx[t])) {
                s_val[t] = ov; s_idx[t] = oi;
            }
            s_sum[t] += s_sum[t + off];
        }
        __syncthreads();
    }

    if (t == 0) {
        const float tscore = p_gen[(size_t)row * VOCAB + tg] * inv + sw * s_sum[0];
        loss_ws[row]  = (tg != PAD_ID) ? -logf(tscore + 1e-12f) : 0.0f;
        out[1 + row]  = (float)s_idx[0];
    }
}

// ---------------- Kernel B: deterministic loss reduction ----------------
__global__ __launch_bounds__(BLK) void loss_reduce_kernel(
    const float* __restrict__ loss_ws, float* __restrict__ out)
{
    __shared__ float s[BLK];
    float a = 0.0f;
    for (int i = threadIdx.x; i < NROWS; i += BLK) a += loss_ws[i];
    s[threadIdx.x] = a;
    __syncthreads();
    for (int off = BLK / 2; off > 0; off >>= 1) {
        if (threadIdx.x < off) s[threadIdx.x] += s[threadIdx.x + off];
        __syncthreads();
    }
    if (threadIdx.x == 0) out[0] = s[0] / (float)NROWS;
}

extern "C" void kernel_launch(void* const* d_in, const int* in_sizes, int n_in,
                              void* d_out, int out_size, void* d_ws, size_t ws_size,
                              hipStream_t stream) {
    const float* p_gen    = (const float*)d_in[0];
    const float* p_copy   = (const float*)d_in[1];
    const float* p_switch = (const float*)d_in[2];
    const int*   tgt      = (const int*)d_in[3];
    const int*   src      = (const int*)d_in[4];
    float* out = (float*)d_out;
    float* ws  = (float*)d_ws;   // NROWS floats of scratch

    ptrgen_row_kernel<<<NROWS, BLK, 0, stream>>>(p_gen, p_copy, p_switch, tgt, src, out, ws);
    loss_reduce_kernel<<<1, BLK, 0, stream>>>(ws, out);
}